// HomoGAT_11914239279716
// MI455X (gfx1250) — compile-verified
//
#include <hip/hip_runtime.h>
#include <hip/hip_bf16.h>
#include <math.h>

// ---------------------------------------------------------------------------
// HomoGAT: 3x (GAT layer -> BatchNorm -> ELU) on gfx1250 (MI455X).
// GEMM uses V_WMMA_F32_16X16X4_F32 (fp32 WMMA, wave32). Edge phase is
// atomic-based segment softmax, fully L2-resident (h/acc = 25.6MB << 192MB L2).
// ---------------------------------------------------------------------------

#define HID 128
#define HEADS 8
#define OUTC 16
#define NEG_SLOPE 0.2f
#define BN_EPS 1e-5f

typedef __attribute__((ext_vector_type(2))) float v2f;
typedef __attribute__((ext_vector_type(8))) float v8f;

// ---------------------------------------------------------------------------
// 1) h = X @ W   (WMMA f32 16x16x4).  Block: 128 rows x 128 cols, 8 waves.
//    Wave w owns N-tile n0 = 16*w (== head w) and loops over 8 M-tiles.
// ---------------------------------------------------------------------------
__global__ __launch_bounds__(256) void gat_gemm_wmma(
    const float* __restrict__ X, const float* __restrict__ W,
    float* __restrict__ H, int nrows) {
  __shared__ float Ws[HID * HID];  // 64 KB, W staged once per block
  const int tid = threadIdx.x;
  for (int i = tid; i < HID * HID; i += 256) Ws[i] = W[i];
  __syncthreads();

  const int wave = tid >> 5;         // 0..7  -> N tile (head)
  const int lane = tid & 31;
  const int half = lane >> 4;        // 0: lanes 0-15 (K=0,1), 1: lanes 16-31 (K=2,3)
  const int l15  = lane & 15;
  const int n0   = wave * 16;
  const int rowBase = blockIdx.x * 128;

  for (int mt = 0; mt < 8; ++mt) {
    const int m0 = rowBase + mt * 16;
    if (m0 >= nrows) break;          // nrows % 16 == 0, tiles fully valid
    v8f c = {};
    const int arow = m0 + l15;
    const float* ap = X + (size_t)arow * HID + half * 2;
    #pragma unroll 4
    for (int kb = 0; kb < HID; kb += 4) {
      v2f a, b;
      a.x = ap[kb];                  // A: 16x4 fp32 fragment (2 VGPRs/lane)
      a.y = ap[kb + 1];
      const int kk = kb + half * 2;  // B: 4x16 fp32 fragment from LDS
      b.x = Ws[kk * HID + n0 + l15];
      b.y = Ws[(kk + 1) * HID + n0 + l15];
      c = __builtin_amdgcn_wmma_f32_16x16x4_f32(
              /*neg_a=*/false, a, /*neg_b=*/false, b,
              /*c_mod=*/(short)0, c, /*reuse_a=*/false, /*reuse_b=*/false);
    }
    // C/D layout: VGPR v -> row m0 + v + 8*half, col n0 + l15
    const int col = n0 + l15;
    const int rb  = m0 + half * 8;
    #pragma unroll
    for (int v = 0; v < 8; ++v)
      H[(size_t)(rb + v) * HID + col] = c[v];
  }
}

// ---------------------------------------------------------------------------
// 2) alpha_s / alpha_d : per (node, head) dot of h-row with attention vectors
// ---------------------------------------------------------------------------
__global__ __launch_bounds__(256) void gat_alpha(
    const float* __restrict__ H, const float* __restrict__ a_src,
    const float* __restrict__ a_dst, float* __restrict__ as,
    float* __restrict__ ad, int n) {
  const int i = blockIdx.x * blockDim.x + threadIdx.x;
  if (i >= n * HEADS) return;
  const int head = i & (HEADS - 1);
  const int node = i >> 3;
  const float* hrow = H + (size_t)node * HID + head * OUTC;
  const float* s = a_src + head * OUTC;
  const float* d = a_dst + head * OUTC;
  float vs = 0.f, vd = 0.f;
  #pragma unroll
  for (int c = 0; c < OUTC; ++c) {
    const float hv = hrow[c];
    vs += hv * s[c];
    vd += hv * d[c];
  }
  as[i] = vs;
  ad[i] = vd;
}

// ---------------------------------------------------------------------------
// 3) init: acc=0, m=-inf, z=0, stats=0
// ---------------------------------------------------------------------------
__global__ __launch_bounds__(256) void gat_init(
    float* __restrict__ acc, float* __restrict__ m, float* __restrict__ z,
    float* __restrict__ stats, int n) {
  const int i = blockIdx.x * blockDim.x + threadIdx.x;
  if (i < n * HID) acc[i] = 0.f;
  if (i < n * HEADS) { m[i] = -INFINITY; z[i] = 0.f; }
  if (i < 2 * HID) stats[i] = 0.f;
}

// float atomic max via int-max / uint-min (valid with -inf init)
__device__ __forceinline__ void atomicMaxF(float* addr, float val) {
  if (val >= 0.f)
    atomicMax((int*)addr, __float_as_int(val));
  else
    atomicMin((unsigned int*)addr, __float_as_uint(val));
}

__device__ __forceinline__ float lrelu(float v) {
  return v >= 0.f ? v : NEG_SLOPE * v;
}

// ---------------------------------------------------------------------------
// 4) segment max over incoming edges
// ---------------------------------------------------------------------------
__global__ __launch_bounds__(256) void gat_edge_max(
    const int* __restrict__ src, const int* __restrict__ dst,
    const float* __restrict__ as, const float* __restrict__ ad,
    float* __restrict__ m, int ne) {
  const int t = blockIdx.x * blockDim.x + threadIdx.x;
  if (t >= ne * HEADS) return;
  const int head = t & (HEADS - 1);
  const int e = t >> 3;
  const int s = src[e], d = dst[e];
  const float v = lrelu(as[s * HEADS + head] + ad[d * HEADS + head]);
  atomicMaxF(&m[d * HEADS + head], v);
}

// ---------------------------------------------------------------------------
// 5) exp(e - m[dst]) -> ebuf; segment sum -> z
// ---------------------------------------------------------------------------
__global__ __launch_bounds__(256) void gat_edge_expsum(
    const int* __restrict__ src, const int* __restrict__ dst,
    const float* __restrict__ as, const float* __restrict__ ad,
    const float* __restrict__ m, float* __restrict__ z,
    float* __restrict__ ebuf, int ne) {
  const int t = blockIdx.x * blockDim.x + threadIdx.x;
  if (t >= ne * HEADS) return;
  const int head = t & (HEADS - 1);
  const int e = t >> 3;
  const int s = src[e], d = dst[e];
  const float v = lrelu(as[s * HEADS + head] + ad[d * HEADS + head]);
  const float ex = expf(v - m[d * HEADS + head]);
  ebuf[t] = ex;
  atomicAdd(&z[d * HEADS + head], ex);
}

// ---------------------------------------------------------------------------
// 6) message pass: acc[dst] += h[src] * alpha  (one thread per edge-channel)
// ---------------------------------------------------------------------------
__global__ __launch_bounds__(256) void gat_edge_msg(
    const int* __restrict__ src, const int* __restrict__ dst,
    const float* __restrict__ H, const float* __restrict__ ebuf,
    const float* __restrict__ z, float* __restrict__ acc, int ne) {
  const int t = blockIdx.x * blockDim.x + threadIdx.x;
  if (t >= ne * HID) return;
  const int ch = t & (HID - 1);
  const int e = t >> 7;
  const int s = src[e], d = dst[e];
  const int head = ch >> 4;  // OUTC == 16
  const float alpha = ebuf[e * HEADS + head] / (z[d * HEADS + head] + 1e-16f);
  atomicAdd(&acc[(size_t)d * HID + ch], H[(size_t)s * HID + ch] * alpha);
}

// ---------------------------------------------------------------------------
// 7) BN stats: per-channel sum & sumsq (LDS partials + global atomics)
//    (bias is a per-channel constant -> cancels exactly in batchnorm; skipped)
// ---------------------------------------------------------------------------
__global__ __launch_bounds__(256) void gat_bn_stats(
    const float* __restrict__ acc, float* __restrict__ stats, int n) {
  __shared__ float ssum[256];
  __shared__ float ssq[256];
  const int ch  = threadIdx.x & (HID - 1);
  const int sub = threadIdx.x >> 7;  // 0/1
  float s = 0.f, q = 0.f;
  for (int r = blockIdx.x * 2 + sub; r < n; r += gridDim.x * 2) {
    const float v = acc[(size_t)r * HID + ch];
    s += v;
    q += v * v;
  }
  ssum[threadIdx.x] = s;
  ssq[threadIdx.x] = q;
  __syncthreads();
  if (sub == 0) {
    atomicAdd(&stats[ch], ssum[threadIdx.x] + ssum[threadIdx.x + 128]);
    atomicAdd(&stats[HID + ch], ssq[threadIdx.x] + ssq[threadIdx.x + 128]);
  }
}

// ---------------------------------------------------------------------------
// 8) BN apply + ELU
// ---------------------------------------------------------------------------
__global__ __launch_bounds__(256) void gat_bn_apply(
    const float* __restrict__ acc, const float* __restrict__ stats,
    const float* __restrict__ gamma, const float* __restrict__ beta,
    float* __restrict__ out, int n) {
  const int i = blockIdx.x * blockDim.x + threadIdx.x;
  if (i >= n * HID) return;
  const int ch = i & (HID - 1);
  const float invn = 1.f / (float)n;
  const float mu = stats[ch] * invn;
  const float var = stats[HID + ch] * invn - mu * mu;
  const float y = (acc[i] - mu) * rsqrtf(var + BN_EPS) * gamma[ch] + beta[ch];
  out[i] = y > 0.f ? y : expf(y) - 1.f;  // ELU(alpha=1)
}

// ---------------------------------------------------------------------------
extern "C" void kernel_launch(void* const* d_in, const int* in_sizes, int n_in,
                              void* d_out, int out_size, void* d_ws, size_t ws_size,
                              hipStream_t stream) {
  const float* x      = (const float*)d_in[0];
  const int*   eidx   = (const int*)d_in[1];   // [2, E] int32 (JAX default x64-off)
  const float* Wall   = (const float*)d_in[2]; // [3,128,128]
  const float* asrc   = (const float*)d_in[3]; // [3,8,16]
  const float* adst   = (const float*)d_in[4]; // [3,8,16]
  const float* gamma  = (const float*)d_in[6]; // [3,128]
  const float* beta   = (const float*)d_in[7]; // [3,128]

  const int N = in_sizes[0] / HID;   // 50000
  const int E = in_sizes[1] / 2;     // 600000
  const int* src = eidx;
  const int* dst = eidx + E;

  float* ws  = (float*)d_ws;
  float* X   = ws;                         // N*128 (layer input ping)
  float* H   = X + (size_t)N * HID;        // N*128 (projected features)
  float* ACC = H + (size_t)N * HID;        // N*128 (aggregated output)
  float* AS  = ACC + (size_t)N * HID;      // N*8
  float* AD  = AS + (size_t)N * HEADS;     // N*8
  float* M   = AD + (size_t)N * HEADS;     // N*8 (segment max)
  float* Z   = M + (size_t)N * HEADS;      // N*8 (segment sum)
  float* ST  = Z + (size_t)N * HEADS;      // 256 (sum, sumsq)
  float* EB  = ST + 2 * HID;               // E*8 (exp values)

  const int gGemm = (N + 127) / 128;
  const int gNH   = (N * HEADS + 255) / 256;
  const int gNC   = (N * HID + 255) / 256;
  const int gEH   = (E * HEADS + 255) / 256;
  const int gEC   = (E * HID + 255) / 256;

  for (int l = 0; l < 3; ++l) {
    const float* xin = (l == 0) ? x : X;
    const float* Wl  = Wall + (size_t)l * HID * HID;
    const float* asl = asrc + l * HEADS * OUTC;
    const float* adl = adst + l * HEADS * OUTC;

    gat_gemm_wmma<<<gGemm, 256, 0, stream>>>(xin, Wl, H, N);
    gat_alpha<<<gNH, 256, 0, stream>>>(H, asl, adl, AS, AD, N);
    gat_init<<<gNC, 256, 0, stream>>>(ACC, M, Z, ST, N);
    gat_edge_max<<<gEH, 256, 0, stream>>>(src, dst, AS, AD, M, E);
    gat_edge_expsum<<<gEH, 256, 0, stream>>>(src, dst, AS, AD, M, Z, EB, E);
    gat_edge_msg<<<gEC, 256, 0, stream>>>(src, dst, H, EB, Z, ACC, E);
    gat_bn_stats<<<512, 256, 0, stream>>>(ACC, ST, N);
    float* outp = (l == 2) ? (float*)d_out : X;
    gat_bn_apply<<<gNC, 256, 0, stream>>>(ACC, ST, gamma + l * HID,
                                          beta + l * HID, outp, N);
  }
}